// LGCNEncoder_43164421324886
// MI455X (gfx1250) — compile-verified
//
#include <hip/hip_runtime.h>

#define USER_COUNT 100000
#define ITEM_COUNT 50000
#define N_NODES    (USER_COUNT + ITEM_COUNT)   // 150000
#define EMB        64
#define N_EDGES    1250000
#define BATCH      4096
#define EDGES_PER_WAVE 4

// ---------------------------------------------------------------------------
// x0 = concat(user_emb, item_emb), vectorized float4 (16B) copies.
// ---------------------------------------------------------------------------
__global__ void lgcn_init_concat(const float* __restrict__ user_emb,
                                 const float* __restrict__ item_emb,
                                 float* __restrict__ x) {
    const int n_user4 = USER_COUNT * EMB / 4;
    const int n_tot4  = N_NODES * EMB / 4;
    int i = blockIdx.x * blockDim.x + threadIdx.x;
    if (i >= n_tot4) return;
    float4 v;
    if (i < n_user4) v = ((const float4*)user_emb)[i];
    else             v = ((const float4*)item_emb)[i - n_user4];
    ((float4*)x)[i] = v;
}

// ---------------------------------------------------------------------------
// Zero the scatter destination (must run every call: ws is not re-poisoned,
// and atomics accumulate into it).
// ---------------------------------------------------------------------------
__global__ void lgcn_zero(float4* __restrict__ y) {
    const int n4 = N_NODES * EMB / 4;
    int i = blockIdx.x * blockDim.x + threadIdx.x;
    if (i >= n4) return;
    y[i] = make_float4(0.f, 0.f, 0.f, 0.f);
}

// ---------------------------------------------------------------------------
// Scatter:  y[row] += w * x[col]  for every edge.
// One wave32 per EDGES_PER_WAVE consecutive edges. Edge metadata is wave-
// uniform: readfirstlane forces it onto the scalar path (s_load_b32, KMcnt),
// so the vector pipe only does the 256B b64 row gather + 2 native f32 L2
// atomics per lane. x and y (38.4 MB each) stay L2-resident (192 MB L2).
// ---------------------------------------------------------------------------
__global__ void lgcn_scatter(const int*   __restrict__ edge_row,
                             const int*   __restrict__ edge_col,
                             const float* __restrict__ edge_w,
                             const float* __restrict__ x,
                             float*       __restrict__ y) {
    const unsigned lane = threadIdx.x & 31u;
    const unsigned waves_per_blk = blockDim.x >> 5;
    // wave_id is uniform across the wave: pin it into an SGPR.
    const unsigned wave_id = __builtin_amdgcn_readfirstlane(
        blockIdx.x * waves_per_blk + (threadIdx.x >> 5));
    const unsigned e0 = wave_id * EDGES_PER_WAVE;

    // Stream edge metadata ahead (speculative global_prefetch_b8).
    if (lane == 0) {
        unsigned pe = e0 + 8192;
        if (pe < N_EDGES) {
            __builtin_prefetch(&edge_col[pe], 0, 0);
            __builtin_prefetch(&edge_row[pe], 0, 0);
            __builtin_prefetch(&edge_w[pe],   0, 0);
        }
    }

#pragma unroll
    for (int k = 0; k < EDGES_PER_WAVE; ++k) {
        const unsigned e = e0 + k;          // scalar (SGPR) index
        if (e >= N_EDGES) return;           // scalar branch
        const int   col = edge_col[e];      // s_load_b32
        const int   row = edge_row[e];      // s_load_b32
        const float w   = edge_w[e];        // s_load_b32

        // SADDR-form b64 gather: scalar base + (col*256 + lane*8) offset.
        const float2 v = *(const float2*)(x + (size_t)col * EMB + lane * 2u);
        float* dst = y + (size_t)row * EMB + lane * 2u;
        // Native GLOBAL_ATOMIC_ADD_F32 (no CAS loop); resolves in L2.
        unsafeAtomicAdd(dst,     w * v.x);
        unsafeAtomicAdd(dst + 1, w * v.y);
    }
}

// ---------------------------------------------------------------------------
// Accumulate 0.25 * x[gathered rows] into d_out.
// accumulate==0: overwrite (first term, also clears harness poison).
// Layout: out rows [0, BATCH) = users, rows [BATCH, 2*BATCH) = items.
// One wave32 per output row, float2 per lane; row index scalarized.
// ---------------------------------------------------------------------------
__global__ void lgcn_gather_acc(const float* __restrict__ x,
                                const int*   __restrict__ users,
                                const int*   __restrict__ items,
                                float*       __restrict__ out,
                                int accumulate) {
    const unsigned lane = threadIdx.x & 31u;
    const unsigned wid = __builtin_amdgcn_readfirstlane(
        blockIdx.x * (blockDim.x >> 5) + (threadIdx.x >> 5));   // 0..8191
    if (wid >= 2u * BATCH) return;

    int node;
    if (wid < BATCH) node = users[wid];                         // s_load_b32
    else             node = USER_COUNT + items[wid - BATCH];    // s_load_b32

    const float2 v = *(const float2*)(x + (size_t)node * EMB + lane * 2u);
    float2* dst = (float2*)(out + (size_t)wid * EMB + lane * 2u);
    float2 r;
    r.x = 0.25f * v.x;
    r.y = 0.25f * v.y;
    if (accumulate) {
        float2 o = *dst;
        r.x += o.x;
        r.y += o.y;
    }
    *dst = r;
}

// ---------------------------------------------------------------------------
extern "C" void kernel_launch(void* const* d_in, const int* in_sizes, int n_in,
                              void* d_out, int out_size, void* d_ws, size_t ws_size,
                              hipStream_t stream) {
    const float* user_emb = (const float*)d_in[0];
    const float* item_emb = (const float*)d_in[1];
    const int*   edge_row = (const int*)d_in[2];
    const int*   edge_col = (const int*)d_in[3];
    const float* edge_w   = (const float*)d_in[4];
    const int*   users    = (const int*)d_in[5];
    const int*   items    = (const int*)d_in[6];
    float*       out      = (float*)d_out;

    // Two ping-pong node buffers: 2 * 150000 * 64 * 4B = 76.8 MB of d_ws.
    float* bufA = (float*)d_ws;
    float* bufB = bufA + (size_t)N_NODES * EMB;

    const int BLK = 256;                                   // 8 waves/block
    const int waves_per_blk = BLK / 32;
    const int grid_node4 = (N_NODES * EMB / 4 + BLK - 1) / BLK;           // 9375
    const int n_waves_scatter =
        (N_EDGES + EDGES_PER_WAVE - 1) / EDGES_PER_WAVE;                  // 312500
    const int grid_scatter =
        (n_waves_scatter + waves_per_blk - 1) / waves_per_blk;            // 39063
    const int grid_gather = (2 * BATCH) / waves_per_blk;                  // 1024

    // x0 = concat(user_emb, item_emb)
    lgcn_init_concat<<<grid_node4, BLK, 0, stream>>>(user_emb, item_emb, bufA);
    // out = 0.25 * x0[rows]   (overwrites poison)
    lgcn_gather_acc<<<grid_gather, BLK, 0, stream>>>(bufA, users, items, out, 0);

    float* x = bufA;
    float* y = bufB;
    for (int layer = 0; layer < 3; ++layer) {
        lgcn_zero<<<grid_node4, BLK, 0, stream>>>((float4*)y);
        lgcn_scatter<<<grid_scatter, BLK, 0, stream>>>(edge_row, edge_col, edge_w, x, y);
        lgcn_gather_acc<<<grid_gather, BLK, 0, stream>>>(y, users, items, out, 1);
        float* t = x; x = y; y = t;
    }
}